// AdvancedHopfieldModel_2731599200530
// MI455X (gfx1250) — compile-verified
//
#include <hip/hip_runtime.h>

// N = 2048 graph energy. Strategy (MI455X):
//  - One fused row pass computes softmax/sigmoid/mask -> x (stored in ws, 16 MB,
//    fits 192 MB L2), row sums (out_flow), and per-row partials (path/binary/edges).
//  - Connectivity: reach recurrence is row-independent, so 10 dependent
//    1x2048 vec-mat products (WMMA f32 16x16x4) replace 10 dense 2048^3 matmuls.
//  - in_flow (column sums) reuses the same WMMA vecmat with r = ones.
//  - All reductions are deterministic (no atomics; fixed tree order).

#define NDIM 2048
#define NN (NDIM * NDIM)
#define KWAVES 8                 // waves per column tile in vecmat (K-split)
#define KCHUNK (NDIM / KWAVES)   // 256 k-values per wave -> 64 WMMA steps

typedef __attribute__((ext_vector_type(2))) float v2f;
typedef __attribute__((ext_vector_type(8))) float v8f;

__device__ __forceinline__ float block_sum(float v, float* smem) {
  const int lane = threadIdx.x & 31;
  const int wid  = threadIdx.x >> 5;
#pragma unroll
  for (int o = 16; o > 0; o >>= 1) v += __shfl_down(v, o, 32);
  if (lane == 0) smem[wid] = v;
  __syncthreads();
  if (wid == 0) {
    float r = (lane < (int)(blockDim.x >> 5)) ? smem[lane] : 0.0f;
#pragma unroll
    for (int o = 16; o > 0; o >>= 1) r += __shfl_down(r, o, 32);
    if (lane == 0) smem[0] = r;
  }
  __syncthreads();
  float r = smem[0];
  __syncthreads();
  return r;
}

__device__ __forceinline__ float block_max(float v, float* smem) {
  const int lane = threadIdx.x & 31;
  const int wid  = threadIdx.x >> 5;
#pragma unroll
  for (int o = 16; o > 0; o >>= 1) v = fmaxf(v, __shfl_down(v, o, 32));
  if (lane == 0) smem[wid] = v;
  __syncthreads();
  if (wid == 0) {
    float r = (lane < (int)(blockDim.x >> 5)) ? smem[lane] : -3.402823466e38f;
#pragma unroll
    for (int o = 16; o > 0; o >>= 1) r = fmaxf(r, __shfl_down(r, o, 32));
    if (lane == 0) smem[0] = r;
  }
  __syncthreads();
  float r = smem[0];
  __syncthreads();
  return r;
}

// One block (256 threads) per row: softmax(attn), x = sigmoid(2*logits)*softmax*valid,
// store x, out_flow[i], and per-row partials (deterministic, no atomics).
__global__ void row_kernel(const float* __restrict__ logits,
                           const float* __restrict__ attn,
                           const float* __restrict__ dist,
                           const float* __restrict__ valid,
                           float* __restrict__ x,
                           float* __restrict__ out_flow,
                           float* __restrict__ row_pc,
                           float* __restrict__ row_bp,
                           float* __restrict__ row_va) {
  __shared__ float smem[32];
  const int i = blockIdx.x;
  const size_t base = (size_t)i * NDIM;

  float av[8];
  float m = -3.402823466e38f;
#pragma unroll
  for (int t = 0; t < 8; ++t) {
    const int j = threadIdx.x + t * 256;
    av[t] = attn[base + j];
    m = fmaxf(m, av[t]);
  }
  m = block_max(m, smem);

  float se = 0.0f;
#pragma unroll
  for (int t = 0; t < 8; ++t) se += expf(av[t] - m);
  se = block_sum(se, smem);
  const float inv = 1.0f / se;

  float sx = 0.0f, pc = 0.0f, bp = 0.0f, va = 0.0f;
#pragma unroll
  for (int t = 0; t < 8; ++t) {
    const int j = threadIdx.x + t * 256;
    const float a = expf(av[t] - m) * inv;           // softmax over axis=1
    const float z = logits[base + j];
    const float s = 1.0f / (1.0f + expf(-2.0f * z)); // sigmoid(logits / 0.5)
    const float v = valid[base + j];
    const float xv = s * a * v;
    x[base + j] = xv;
    sx += xv;
    bp += xv * (1.0f - xv);
    va += v;
    pc += dist[base + j] * xv;                       // xv==0 kills 1e9 sentinels
  }
  sx = block_sum(sx, smem);
  pc = block_sum(pc, smem);
  bp = block_sum(bp, smem);
  va = block_sum(va, smem);
  if (threadIdx.x == 0) {
    out_flow[i] = sx;
    row_pc[i] = pc;
    row_bp[i] = bp;
    row_va[i] = va;
  }
}

// Deterministic second-stage reduction of per-row partials.
// scal: [0]=path_cost [1]=binary [2]=sum_x [3]=n_edges [4]=flow_penalty
__global__ void reduce_kernel(const float* __restrict__ out_flow,
                              const float* __restrict__ row_pc,
                              const float* __restrict__ row_bp,
                              const float* __restrict__ row_va,
                              float* __restrict__ scal) {
  __shared__ float smem[32];
  float a = 0.0f, b = 0.0f, c = 0.0f, d = 0.0f;
  for (int j = threadIdx.x; j < NDIM; j += 256) {
    a += row_pc[j];
    b += row_bp[j];
    c += out_flow[j];
    d += row_va[j];
  }
  a = block_sum(a, smem);
  b = block_sum(b, smem);
  c = block_sum(c, smem);
  d = block_sum(d, smem);
  if (threadIdx.x == 0) {
    scal[0] = a; scal[1] = b; scal[2] = c; scal[3] = d;
  }
}

__global__ void flow_kernel(const float* __restrict__ out_flow,
                            const float* __restrict__ in_flow,
                            const int* __restrict__ srcp,
                            const int* __restrict__ dstp,
                            float* __restrict__ scal) {
  __shared__ float smem[32];
  const int src = *srcp;
  const int dst = *dstp;
  float s = 0.0f;
  for (int i = threadIdx.x; i < NDIM; i += 256) {
    float d = out_flow[i] - in_flow[i];
    if (i == src) d -= 1.0f;
    if (i == dst) d += 1.0f;
    s += d * d;
  }
  s = block_sum(s, smem);
  if (threadIdx.x == 0) scal[4] = s;
}

__global__ void ones_kernel(float* __restrict__ ones) {
  const int j = blockIdx.x * 256 + threadIdx.x;
  if (j < NDIM) ones[j] = 1.0f;
}

__global__ void copyrow_kernel(const float* __restrict__ x,
                               const int* __restrict__ srcp,
                               float* __restrict__ r) {
  const int j = blockIdx.x * 256 + threadIdx.x;
  if (j < NDIM) r[j] = x[(size_t)(*srcp) * NDIM + j];
}

// t = r @ x  (1x2048 times 2048x2048) via V_WMMA_F32_16X16X4_F32.
// Layouts (16x16x4 f32): A lane m+16*(k>>1) vgpr k&1 holds A[m][k];
// B lane n+16*? : lane l holds column n=l&15, K = vgpr + 2*(l>>4);
// C row M=0 lives in VGPR0 of lanes 0..15.
// Trick: rows of D are independent, and we only read row 0 — so every lane
// broadcasts the same r chunk into A (no zeroing, no lane selects, no EXEC
// divergence); all 16 rows of D compute the same exact result.
// K is split across the block's 8 waves (one column tile per block); partial
// row-0 results are combined through LDS in a fixed order (deterministic).
// mode 0: out = t (column sums when r = ones). mode 1: out = min(r + t, 1).
__global__ void vecmat_kernel(const float* __restrict__ r,
                              const float* __restrict__ x,
                              float* __restrict__ out,
                              int mode) {
  __shared__ float sr[NDIM];
  __shared__ float part[KWAVES * 16];
  for (int t = threadIdx.x; t < NDIM; t += 256) sr[t] = r[t];
  __syncthreads();

  const int lane = threadIdx.x & 31;
  const int wave = threadIdx.x >> 5;        // 0..7 -> K chunk
  const int j0 = blockIdx.x * 16;           // one 16-column tile per block
  const int half = lane >> 4;               // 0 or 1
  const int jn = j0 + (lane & 15);
  const int k0 = wave * KCHUNK;

  v8f c = {};
#pragma unroll 8
  for (int k = k0; k < k0 + KCHUNK; k += 4) {
    const int krow = k + 2 * half;          // lanes 0-15: k ; lanes 16-31: k+2
    const size_t rowb = (size_t)krow * NDIM + jn;
    v2f b;
    b[0] = x[rowb];                         // B[K=krow  ][n]
    b[1] = x[rowb + NDIM];                  // B[K=krow+1][n]
    v2f a;
    a[0] = sr[krow];                        // broadcast r chunk to all rows of A
    a[1] = sr[krow + 1];
    c = __builtin_amdgcn_wmma_f32_16x16x4_f32(false, a, false, b,
                                              (short)0, c, false, false);
  }

  if (lane < 16) part[wave * 16 + lane] = c[0];   // row M=0 of this wave's D
  __syncthreads();
  if (threadIdx.x < 16) {
    float t = 0.0f;
#pragma unroll
    for (int w = 0; w < KWAVES; ++w) t += part[w * 16 + threadIdx.x];
    const int j = j0 + threadIdx.x;
    out[j] = (mode == 0) ? t : fminf(sr[j] + t, 1.0f);
  }
}

__global__ void finalize_kernel(const float* __restrict__ scal,
                                const float* __restrict__ rfin,
                                const int* __restrict__ dstp,
                                float* __restrict__ out) {
  const float path  = scal[0];
  const float bin   = scal[1];
  const float sumx  = scal[2];
  const float edges = scal[3];
  const float flow  = scal[4];
  const float density = edges / (float)NN;
  const float mu2 = 10.0f * (1.0f + density);   // mu3 == mu2
  const float reach_sd = rfin[*dstp];
  const float cd = 1.0f - reach_sd;
  const float energy = path / (edges + 1e-6f)
                     + mu2 * flow / (float)NDIM
                     + mu2 * bin / (float)NN
                     + 20.0f * cd * cd
                     + 5.0f * sumx / (float)NN;
  out[0] = energy;
}

extern "C" void kernel_launch(void* const* d_in, const int* in_sizes, int n_in,
                              void* d_out, int out_size, void* d_ws, size_t ws_size,
                              hipStream_t stream) {
  const float* logits = (const float*)d_in[0];
  const float* attn   = (const float*)d_in[1];
  const float* dist   = (const float*)d_in[2];
  const float* valid  = (const float*)d_in[3];
  const int*   srcp   = (const int*)d_in[4];
  const int*   dstp   = (const int*)d_in[5];

  float* ws       = (float*)d_ws;
  float* x        = ws;                       // N*N
  float* out_flow = ws + (size_t)NN;          // N
  float* in_flow  = out_flow + NDIM;          // N
  float* ones     = in_flow + NDIM;           // N
  float* r0       = ones + NDIM;              // N
  float* r1       = r0 + NDIM;                // N
  float* row_pc   = r1 + NDIM;                // N
  float* row_bp   = row_pc + NDIM;            // N
  float* row_va   = row_bp + NDIM;            // N
  float* scal     = row_va + NDIM;            // 8

  ones_kernel<<<NDIM / 256, 256, 0, stream>>>(ones);
  row_kernel<<<NDIM, 256, 0, stream>>>(logits, attn, dist, valid, x,
                                       out_flow, row_pc, row_bp, row_va);
  // in_flow = ones @ x (column sums) via WMMA vecmat
  vecmat_kernel<<<NDIM / 16, 256, 0, stream>>>(ones, x, in_flow, 0);
  reduce_kernel<<<1, 256, 0, stream>>>(out_flow, row_pc, row_bp, row_va, scal);
  flow_kernel<<<1, 256, 0, stream>>>(out_flow, in_flow, srcp, dstp, scal);

  // connectivity: r = x[source,:]; 10x r = min(r + r@x, 1); final lands in r0
  copyrow_kernel<<<NDIM / 256, 256, 0, stream>>>(x, srcp, r0);
  float* bufs[2] = { r0, r1 };
  for (int it = 0; it < 10; ++it) {
    vecmat_kernel<<<NDIM / 16, 256, 0, stream>>>(bufs[it & 1], x,
                                                 bufs[(it + 1) & 1], 1);
  }

  finalize_kernel<<<1, 1, 0, stream>>>(scal, r0, dstp, (float*)d_out);
}